// BinaryMNModel_5540507812481
// MI455X (gfx1250) — compile-verified
//
#include <hip/hip_runtime.h>

// Problem constants (reference: B=256, V=5000, C=20000, S=3, NC=8)
#define B_N   256
#define V_N   5000
#define C_N   20000
#define CSPLIT 32
#define CCHUNK (C_N / CSPLIT)          // 625 clauses per block
#define BTILE  16
#define TILE   160                     // clauses per LDS tile (5 wave-iters)
#define NTILE  ((CCHUNK + TILE - 1) / TILE)   // 4 tiles per chunk

typedef float        v2f   __attribute__((ext_vector_type(2)));
typedef float        v8f   __attribute__((ext_vector_type(8)));
typedef unsigned int u32x4 __attribute__((ext_vector_type(4)));
typedef int          i32x4 __attribute__((ext_vector_type(4)));
typedef int          i32x8 __attribute__((ext_vector_type(8)));

// Generic LDS pointer -> byte offset in LDS space (flat->LDS uses addr[31:0]).
__device__ __forceinline__ unsigned int lds_off(const void* p) {
    return (unsigned int)(uintptr_t)p;
}

// 1D Tensor-Data-Mover copy: n_tile 4-byte elements from gptr to the LDS byte offset in lds_byte.
// n_rem = elements from gptr to end of the source array: descriptor tensor_dim0,
// so tail-tile overreads are TDM OOB reads (return zero) instead of faults.
// D# layout per CDNA5 ISA ch. 8 (group0 128b, group1 256b, groups 2/3 zero).
__device__ __forceinline__ void tdm_load_1d(unsigned int lds_byte,
                                            const void* gptr,
                                            unsigned int n_tile,
                                            unsigned int n_rem) {
    const unsigned long long ga = (unsigned long long)(uintptr_t)gptr;
    u32x4 g0;
    g0[0] = 1u;                                        // count=1, user desc
    g0[1] = lds_byte;                                  // lds_addr
    g0[2] = (unsigned int)(ga & 0xFFFFFFFFu);          // global_addr[31:0]
    g0[3] = (unsigned int)((ga >> 32) & 0x1FFFFFFu)    // global_addr[56:32]
          | (2u << 30);                                // type = 2 ("image")
    i32x8 g1;
    g1[0] = 0x20000;                                   // data_size=2 -> 4 bytes
    g1[1] = (int)((n_rem & 0xFFFFu) << 16);            // tensor_dim0[15:0]
    g1[2] = (int)(((n_rem >> 16) & 0xFFFFu)            // tensor_dim0[31:16]
          | (1u << 16));                               // tensor_dim1 = 1
    g1[3] = (int)((n_tile & 0xFFFFu) << 16);           // tile_dim0
    g1[4] = 1;                                         // tile_dim1=1, tile_dim2=0
    g1[5] = (int)n_rem;                                // tensor_dim0_stride[31:0]
    g1[6] = 0;                                         // stride0 hi / stride1 lo
    g1[7] = 0;                                         // stride1 hi
    const i32x4 gz = {0, 0, 0, 0};                     // groups 2/3: dims unused
#if defined(__clang_major__) && (__clang_major__ >= 23)
    const i32x8 gz8 = {0, 0, 0, 0, 0, 0, 0, 0};
    __builtin_amdgcn_tensor_load_to_lds(g0, g1, gz, gz, gz8, 0);
#else
    __builtin_amdgcn_tensor_load_to_lds(g0, g1, gz, gz, 0);
#endif
}

// Score of one clause (clause data from LDS, x gathered from global/L2):
//   sum_{j=0..7} f[j] * sel(a0,j>>2&1)*sel(a1,j>>1&1)*sel(a2,j&1),
// sel(a,1)=a, sel(a,0)=1-a. Factored: 15 VALU ops.
__device__ __forceinline__ float clause_score(const float* __restrict__ xrow,
                                              const int* sv, const float* sf,
                                              int lc) {
    const int v0 = sv[lc * 3 + 0];
    const int v1 = sv[lc * 3 + 1];
    const int v2 = sv[lc * 3 + 2];
    const float4 fA = reinterpret_cast<const float4*>(sf)[lc * 2 + 0]; // f0..f3
    const float4 fB = reinterpret_cast<const float4*>(sf)[lc * 2 + 1]; // f4..f7
    const float a0 = xrow[v0], a1 = xrow[v1], a2 = xrow[v2];
    const float n0 = 1.0f - a0, n1 = 1.0f - a1, n2 = 1.0f - a2;
    const float m00 = n0 * n1, m01 = n0 * a1, m10 = a0 * n1, m11 = a0 * a1;
    const float ev = fA.x * m00 + fA.z * m01 + fB.x * m10 + fB.z * m11;
    const float od = fA.y * m00 + fA.w * m01 + fB.y * m10 + fB.w * m11;
    return n2 * ev + a2 * od;
}

__global__ __launch_bounds__(256) void bmn_main(const float* __restrict__ x,
                                                const int* __restrict__ vars,
                                                const float* __restrict__ factors,
                                                float* __restrict__ partial) {
    __shared__ struct __align__(16) {
        int   sv[2][TILE * 3];     // clause var indices, double-buffered
        float sf[2][TILE * 8];     // clause factors, double-buffered
        float wave_tot[8][16];     // per-wave reduction scratch
    } sm;

    const int tid  = threadIdx.x;
    const int w    = tid >> 5;          // wave id, 0..7
    const int lane = tid & 31;
    const int bi   = lane & 15;         // batch row within tile
    const int h    = lane >> 4;         // half-wave clause slot
    const int bt   = blockIdx.x & 15;   // batch tile
    const int ct   = blockIdx.x >> 4;   // clause chunk
    const int b0   = bt * BTILE;
    const int cb   = ct * CCHUNK;
    const int ce   = cb + CCHUNK;

    const float* xrow = x + (size_t)(b0 + bi) * V_N;

    // Prologue: wave 0 DMAs tile 0 into buffer 0 and waits on TENSORcnt.
    if (w == 0) {
        tdm_load_1d(lds_off(&sm.sv[0][0]), vars    + (size_t)cb * 3,
                    TILE * 3, (unsigned int)(C_N - cb) * 3);
        tdm_load_1d(lds_off(&sm.sf[0][0]), factors + (size_t)cb * 8,
                    TILE * 8, (unsigned int)(C_N - cb) * 8);
        __builtin_amdgcn_s_wait_tensorcnt(0);
    }
    __syncthreads();

    // 4 f32 partials per batch row per wave (2 accs x 2 half-wave slots)
    // = the K=4 of V_WMMA_F32_16X16X4_F32.
    float acc0 = 0.0f, acc1 = 0.0f;

    for (int t = 0; t < NTILE; ++t) {
        const int p = t & 1;
        // Issue next tile's DMA into the other buffer before computing.
        if (w == 0 && (t + 1) < NTILE) {
            const int tb1 = cb + (t + 1) * TILE;
            tdm_load_1d(lds_off(&sm.sv[1 - p][0]), vars    + (size_t)tb1 * 3,
                        TILE * 3, (unsigned int)(C_N - tb1) * 3);
            tdm_load_1d(lds_off(&sm.sf[1 - p][0]), factors + (size_t)tb1 * 8,
                        TILE * 8, (unsigned int)(C_N - tb1) * 8);
        }
        const int tb = cb + t * TILE;
        const int* sv = &sm.sv[p][0];
        const float* sf = &sm.sf[p][0];
#pragma unroll
        for (int lc4 = w * 4; lc4 < TILE; lc4 += 32) {
            const int lc0 = lc4 + h;
            const int lc1 = lc4 + 2 + h;
            if (tb + lc0 < ce) acc0 += clause_score(xrow, sv, sf, lc0);
            if (tb + lc1 < ce) acc1 += clause_score(xrow, sv, sf, lc1);
        }
        __syncthreads();                 // everyone done with buffer p
        if (w == 0 && (t + 1) < NTILE)
            __builtin_amdgcn_s_wait_tensorcnt(0);   // next buffer landed
        __syncthreads();                 // release buffer 1-p
    }

    // Collapse the 4 partials per batch row with one f32 WMMA:
    //   A (16x4 f32): lanes 0-15 carry K=0,1 (acc0,acc1); lanes 16-31 K=2,3.
    //   B = all-ones (4x16)  ->  D[m,n] = total of row m, any n.
    v2f A;  A[0] = acc0;  A[1] = acc1;
    v2f Bm; Bm[0] = 1.0f; Bm[1] = 1.0f;
    v8f Cm = {};
    v8f D = __builtin_amdgcn_wmma_f32_16x16x4_f32(
        false, A, false, Bm, (short)0, Cm, false, false);

    // D layout: VGPR v, lanes 0-15 -> M=v; lanes 16-31 -> M=v+8 (N=0 at lane 0/16).
    if (lane == 0) {
#pragma unroll
        for (int v = 0; v < 8; ++v) sm.wave_tot[w][v] = D[v];
    } else if (lane == 16) {
#pragma unroll
        for (int v = 0; v < 8; ++v) sm.wave_tot[w][8 + v] = D[v];
    }
    __syncthreads();

    // Deterministic cross-wave sum (fixed order), one partial per (chunk, b).
    if (tid < 16) {
        float s = 0.0f;
#pragma unroll
        for (int ww = 0; ww < 8; ++ww) s += sm.wave_tot[ww][tid];
        partial[(size_t)ct * B_N + b0 + tid] = s;
    }
}

__global__ __launch_bounds__(256) void bmn_reduce(const float* __restrict__ partial,
                                                  float* __restrict__ out) {
    const int b = threadIdx.x;
    float s = 0.0f;
#pragma unroll
    for (int ct = 0; ct < CSPLIT; ++ct) s += partial[(size_t)ct * B_N + b];
    out[b] = s;
}

extern "C" void kernel_launch(void* const* d_in, const int* in_sizes, int n_in,
                              void* d_out, int out_size, void* d_ws, size_t ws_size,
                              hipStream_t stream) {
    (void)in_sizes; (void)n_in; (void)out_size; (void)ws_size;
    const float* x       = (const float*)d_in[0];
    // d_in[1] = binary_combinations: fixed bit-pattern matrix; its structure
    // (bit k of j selects a vs 1-a for var S-1-k) is hardcoded in clause_score.
    const int*   vars    = (const int*)d_in[2];
    const float* factors = (const float*)d_in[3];
    float* partial = (float*)d_ws;               // CSPLIT * 256 floats = 32 KB

    bmn_main<<<dim3(16 * CSPLIT), dim3(256), 0, stream>>>(x, vars, factors, partial);
    bmn_reduce<<<dim3(1), dim3(B_N), 0, stream>>>(partial, (float*)d_out);
}